// HEATConv_52604759441969
// MI455X (gfx1250) — compile-verified
//
#include <hip/hip_runtime.h>
#include <hip/hip_bf16.h>

#define N_NODES 50000
#define N_EDGES 400000
#define SLOPE 0.2f

typedef __attribute__((ext_vector_type(2))) float v2f;
typedef __attribute__((ext_vector_type(8))) float v8f;

__device__ __forceinline__ v8f wmma4(v2f a, v2f b, v8f c) {
    // V_WMMA_F32_16X16X4_F32 : D = A(16x4) * B(4x16) + C(16x16)
    return __builtin_amdgcn_wmma_f32_16x16x4_f32(false, a, false, b, (short)0, c, false, false);
}

__device__ __forceinline__ float lrelu(float x) { return x >= 0.f ? x : SLOPE * x; }

// monotonic float->uint encoding for atomicMax-based segment max
__device__ __forceinline__ unsigned fenc(float f) {
    unsigned b = __float_as_uint(f);
    return (b & 0x80000000u) ? ~b : (b | 0x80000000u);
}
__device__ __forceinline__ float fdec(unsigned u) {
    unsigned b = (u & 0x80000000u) ? (u & 0x7FFFFFFFu) : ~u;
    return __uint_as_float(b);
}

// ---------------------------------------------------------------------------
// init: zero out[N,1024], denom[N,8]; segmax[N,8] = 0 (== encoding below any real)
// ---------------------------------------------------------------------------
__global__ void k_init(float* __restrict__ out, float* __restrict__ denom,
                       unsigned* __restrict__ segmax) {
    size_t i = (size_t)blockIdx.x * blockDim.x + threadIdx.x;
    if (i < (size_t)N_NODES * 1024) out[i] = 0.f;
    if (i < (size_t)N_NODES * 8) { denom[i] = 0.f; segmax[i] = 0u; }
}

// ---------------------------------------------------------------------------
// Kernel 1: node_emb[n,128] = node_feats[n,:256] @ W_node[type[n]]
//           s_node[n,0:8]  = node_emb[n] @ W_attn[0:128]     (tar chunk)
//           s_node[n,8:16] = node_emb[n] @ W_attn[384:512]   (src chunk)
// block = 256 threads (8 waves), 16 nodes per block.
// ---------------------------------------------------------------------------
__global__ __launch_bounds__(256) void k_node_emb(
    const float* __restrict__ node_feats, const int* __restrict__ node_type,
    const float* __restrict__ W_node, const float* __restrict__ W_attn,
    float* __restrict__ node_emb, float* __restrict__ s_node)
{
    __shared__ float feats[16 * 256];
    __shared__ float embt[16 * 128];
    const int nbase = blockIdx.x * 16;
    const int tid = threadIdx.x;

    // stage node_feats tile [16][256] (1024 float4, coalesced)
    for (int i = tid; i < 1024; i += 256) {
        int row = i >> 6, c4 = i & 63;
        ((float4*)feats)[i] =
            ((const float4*)(node_feats + (size_t)(nbase + row) * 256))[c4];
    }
    // prefetch next block's node_feats tile toward GL2 (16KB = 128 lines)
    if (tid < 128 && nbase + 16 < N_NODES)
        __builtin_prefetch(node_feats + (size_t)(nbase + 16) * 256 + tid * 32, 0, 1);
    __syncthreads();

    const int lane = tid & 31, wave = tid >> 5;
    const int n = lane & 15;                 // N-col / A-row within tile
    const int khalf = (lane < 16) ? 0 : 2;   // K sub-offset per half-wave
    const int col = wave * 16 + n;           // output column 0..127
    const float* frow = feats + n * 256;

    v8f acc0 = {}, acc1 = {}, acc2 = {};
    for (int k = 0; k < 256; k += 4) {
        const int kk = k + khalf;
        v2f a; a.x = frow[kk]; a.y = frow[kk + 1];
        const float* wb = W_node + (size_t)kk * 128 + col;
        v2f b0, b1, b2;
        b0.x = wb[0];              b0.y = wb[128];
        b1.x = wb[256 * 128];      b1.y = wb[256 * 128 + 128];
        b2.x = wb[2 * 256 * 128];  b2.y = wb[2 * 256 * 128 + 128];
        acc0 = wmma4(a, b0, acc0);
        acc1 = wmma4(a, b1, acc1);
        acc2 = wmma4(a, b2, acc2);
    }

    // per-row type select (reference computes all T then gathers) + store
    #pragma unroll
    for (int r = 0; r < 8; ++r) {
        int M = r + ((lane < 16) ? 0 : 8);
        int t = node_type[nbase + M];
        float v = (t == 0) ? acc0[r] : (t == 1) ? acc1[r] : acc2[r];
        node_emb[(size_t)(nbase + M) * 128 + col] = v;
        embt[M * 128 + col] = v;
    }
    __syncthreads();

    // second GEMM: [16,128] @ [128,16] -> s_node (cols 0-7 tar, 8-15 src)
    if (wave == 0) {
        v8f c = {};
        const float* erow = embt + n * 128;
        for (int k = 0; k < 128; k += 4) {
            const int kk = k + khalf;
            v2f a; a.x = erow[kk]; a.y = erow[kk + 1];
            v2f b;
            if (n < 8) { b.x = W_attn[kk * 8 + n];          b.y = W_attn[(kk + 1) * 8 + n]; }
            else       { b.x = W_attn[(384 + kk) * 8 + n - 8]; b.y = W_attn[(385 + kk) * 8 + n - 8]; }
            c = wmma4(a, b, c);
        }
        #pragma unroll
        for (int r = 0; r < 8; ++r) {
            int M = r + ((lane < 16) ? 0 : 8);
            s_node[(size_t)(nbase + M) * 16 + n] = c[r];
        }
    }
}

// ---------------------------------------------------------------------------
// Kernel 2: (y==0) attr_emb = lrelu(edge_attr @ W_eattr), stored; plus
//           e_sc_attr[e,8] = attr_emb[e] @ W_attn[128:256]
//           (y==1) type path, type_emb never hits global memory:
//           e_sc_type[e,8] = lrelu(edge_type @ W_etype) @ W_attn[256:384]
// ---------------------------------------------------------------------------
__global__ __launch_bounds__(256) void k_edge_emb(
    const float* __restrict__ edge_attr, const float* __restrict__ edge_type,
    const float* __restrict__ W_eattr, const float* __restrict__ W_etype,
    const float* __restrict__ W_attn, float* __restrict__ attr_emb,
    float* __restrict__ e_sc_attr, float* __restrict__ e_sc_type)
{
    __shared__ float intile[16 * 64];
    __shared__ float embt[16 * 128];
    const int ebase = blockIdx.x * 16;
    const int variant = blockIdx.y;
    const float* in = variant ? edge_type : edge_attr;
    const float* W  = variant ? W_etype : W_eattr;
    const int tid = threadIdx.x;

    { // stage [16][64] tile: 256 float4, one per thread
        int row = tid >> 4, c4 = tid & 15;
        ((float4*)intile)[tid] =
            ((const float4*)(in + (size_t)(ebase + row) * 64))[c4];
    }
    // prefetch next block's streaming input tile (4KB = 32 lines)
    if (tid < 32 && ebase + 16 < N_EDGES)
        __builtin_prefetch(in + (size_t)(ebase + 16) * 64 + tid * 32, 0, 1);
    __syncthreads();

    const int lane = tid & 31, wave = tid >> 5;
    const int n = lane & 15;
    const int khalf = (lane < 16) ? 0 : 2;
    const int col = wave * 16 + n;
    const float* frow = intile + n * 64;

    v8f acc = {};
    for (int k = 0; k < 64; k += 4) {
        const int kk = k + khalf;
        v2f a; a.x = frow[kk]; a.y = frow[kk + 1];
        const float* wb = W + (size_t)kk * 128 + col;
        v2f b; b.x = wb[0]; b.y = wb[128];
        acc = wmma4(a, b, acc);
    }
    #pragma unroll
    for (int r = 0; r < 8; ++r) {
        int M = r + ((lane < 16) ? 0 : 8);
        float v = lrelu(acc[r]);
        embt[M * 128 + col] = v;
        if (!variant) attr_emb[(size_t)(ebase + M) * 128 + col] = v;
    }
    __syncthreads();

    if (wave == 0) {
        v8f c = {};
        const float* erow = embt + n * 128;
        const float* Wa = W_attn + (variant ? 256 * 8 : 128 * 8);
        for (int k = 0; k < 128; k += 4) {
            const int kk = k + khalf;
            v2f a; a.x = erow[kk]; a.y = erow[kk + 1];
            v2f b;
            if (n < 8) { b.x = Wa[kk * 8 + n]; b.y = Wa[(kk + 1) * 8 + n]; }
            else       { b.x = 0.f; b.y = 0.f; }
            c = wmma4(a, b, c);
        }
        float* dst = variant ? e_sc_type : e_sc_attr;
        if (n < 8) {
            #pragma unroll
            for (int r = 0; r < 8; ++r) {
                int M = r + ((lane < 16) ? 0 : 8);
                dst[(size_t)(ebase + M) * 8 + n] = c[r];
            }
        }
    }
}

// ---------------------------------------------------------------------------
// scores[e,h] = lrelu(e_sc_attr + e_sc_type + s_node[tar].tar + s_node[src].src)
// + segment max over src via encoded atomicMax
// ---------------------------------------------------------------------------
__global__ void k_scores(const int* __restrict__ src, const int* __restrict__ tar,
                         const float* __restrict__ e_sc_attr,
                         const float* __restrict__ e_sc_type,
                         const float* __restrict__ s_node,
                         float* __restrict__ scores, unsigned* __restrict__ segmax)
{
    int i = blockIdx.x * blockDim.x + threadIdx.x;
    if (i >= N_EDGES * 8) return;
    int e = i >> 3, h = i & 7;
    int s = src[e], t = tar[e];
    float sc = e_sc_attr[i] + e_sc_type[i]
             + s_node[(size_t)t * 16 + h] + s_node[(size_t)s * 16 + 8 + h];
    sc = lrelu(sc);
    scores[i] = sc;
    atomicMax(&segmax[(size_t)s * 8 + h], fenc(sc));
}

__global__ void k_expsum(const int* __restrict__ src, float* __restrict__ scores,
                         const unsigned* __restrict__ segmax, float* __restrict__ denom)
{
    int i = blockIdx.x * blockDim.x + threadIdx.x;
    if (i >= N_EDGES * 8) return;
    int e = i >> 3, h = i & 7;
    int s = src[e];
    float m = fdec(segmax[(size_t)s * 8 + h]);
    float ex = __expf(scores[i] - m);
    scores[i] = ex;   // reuse buffer: now holds exp()
    atomicAdd(&denom[(size_t)s * 8 + h], ex);
}

// ---------------------------------------------------------------------------
// Kernel 5: msg = lrelu([node_emb[src[e]] | attr_emb[e]] @ W_upd) ; then
//           out[src[e], h, :] += attn[e,h] * msg[e,:]  (atomic scatter)
// ---------------------------------------------------------------------------
__global__ __launch_bounds__(256) void k_msg_scatter(
    const int* __restrict__ src, const float* __restrict__ node_emb,
    const float* __restrict__ attr_emb, const float* __restrict__ W_upd,
    const float* __restrict__ ex, const float* __restrict__ denom,
    float* __restrict__ out)
{
    __shared__ float afeat[16 * 256];
    __shared__ float attn[16 * 8];
    __shared__ int srcs[16];
    const int ebase = blockIdx.x * 16;
    const int tid = threadIdx.x;

    if (tid < 16) srcs[tid] = src[ebase + tid];
    __syncthreads();

    // stage [16][256]: cols 0-127 gathered node_emb rows (L2-resident, 25.6MB),
    // cols 128-255 streaming attr_emb rows
    for (int i = tid; i < 1024; i += 256) {
        int row = i >> 6, c4 = i & 63;
        float4 v;
        if (c4 < 32)
            v = ((const float4*)(node_emb + (size_t)srcs[row] * 128))[c4];
        else
            v = ((const float4*)(attr_emb + (size_t)(ebase + row) * 128))[c4 - 32];
        ((float4*)afeat)[i] = v;
    }
    // prefetch next block's streaming attr_emb tile (8KB = 64 lines)
    if (tid < 64 && ebase + 16 < N_EDGES)
        __builtin_prefetch(attr_emb + (size_t)(ebase + 16) * 128 + tid * 32, 0, 1);
    if (tid < 128) {
        int row = tid >> 3, h = tid & 7;
        attn[tid] = ex[(size_t)(ebase + row) * 8 + h]
                  / denom[(size_t)srcs[row] * 8 + h];
    }
    __syncthreads();

    const int lane = tid & 31, wave = tid >> 5;
    const int n = lane & 15;
    const int khalf = (lane < 16) ? 0 : 2;
    const int col = wave * 16 + n;
    const float* frow = afeat + n * 256;

    v8f acc = {};
    for (int k = 0; k < 256; k += 4) {
        const int kk = k + khalf;
        v2f a; a.x = frow[kk]; a.y = frow[kk + 1];
        const float* wb = W_upd + (size_t)kk * 128 + col;
        v2f b; b.x = wb[0]; b.y = wb[128];
        acc = wmma4(a, b, acc);
    }
    #pragma unroll
    for (int r = 0; r < 8; ++r) {
        int M = r + ((lane < 16) ? 0 : 8);
        float v = lrelu(acc[r]);
        float* obase = out + (size_t)srcs[M] * 1024 + col;
        const float* arow = attn + M * 8;
        #pragma unroll
        for (int h = 0; h < 8; ++h)
            atomicAdd(obase + h * 128, arow[h] * v);
    }
}

// ---------------------------------------------------------------------------
extern "C" void kernel_launch(void* const* d_in, const int* in_sizes, int n_in,
                              void* d_out, int out_size, void* d_ws, size_t ws_size,
                              hipStream_t stream) {
    const float* node_feats = (const float*)d_in[0];
    const int*   edge_index = (const int*)d_in[1];   // [2,E]
    const float* edge_attr  = (const float*)d_in[2];
    const float* edge_type  = (const float*)d_in[3];
    const int*   node_type  = (const int*)d_in[4];
    // d_in[5] edge_type_list: unused by reference math
    const float* W_node  = (const float*)d_in[6];
    const float* W_eattr = (const float*)d_in[7];
    const float* W_etype = (const float*)d_in[8];
    const float* W_upd   = (const float*)d_in[9];
    const float* W_attn  = (const float*)d_in[10];
    float* out = (float*)d_out;

    const int* src = edge_index;            // edge_index[0]
    const int* tar = edge_index + N_EDGES;  // edge_index[1]

    // workspace carve-up (bytes)
    char* ws = (char*)d_ws;
    float*    node_emb  = (float*)(ws);                       // N*128
    float*    attr_emb  = (float*)(ws + 25600000);            // E*128
    float*    s_node    = (float*)(ws + 230400000);           // N*16
    float*    e_sc_attr = (float*)(ws + 233600000);           // E*8
    float*    e_sc_type = (float*)(ws + 246400000);           // E*8
    float*    scores    = (float*)(ws + 259200000);           // E*8 (later exp)
    unsigned* segmax    = (unsigned*)(ws + 272000000);        // N*8
    float*    denom     = (float*)(ws + 273600000);           // N*8
    // total 275,200,000 bytes

    k_init<<<(N_NODES * 1024 + 255) / 256, 256, 0, stream>>>(out, denom, segmax);

    k_node_emb<<<N_NODES / 16, 256, 0, stream>>>(node_feats, node_type, W_node,
                                                 W_attn, node_emb, s_node);

    dim3 g2(N_EDGES / 16, 2);
    k_edge_emb<<<g2, 256, 0, stream>>>(edge_attr, edge_type, W_eattr, W_etype,
                                       W_attn, attr_emb, e_sc_attr, e_sc_type);

    k_scores<<<(N_EDGES * 8 + 255) / 256, 256, 0, stream>>>(src, tar, e_sc_attr,
                                                            e_sc_type, s_node,
                                                            scores, segmax);

    k_expsum<<<(N_EDGES * 8 + 255) / 256, 256, 0, stream>>>(src, scores, segmax,
                                                            denom);

    k_msg_scatter<<<N_EDGES / 16, 256, 0, stream>>>(src, node_emb, attr_emb,
                                                    W_upd, scores, denom, out);
}